// MoERouter_47863115547007
// MI455X (gfx1250) — compile-verified
//
#include <hip/hip_runtime.h>

#define TOKENS 32768
#define HIDDEN 2048
#define NEXP   64
#define MTILE  128
#define KCHUNK 64
#define NCHUNK (HIDDEN / KCHUNK)
#define PITCH  68   // floats; 272B rows -> 16B aligned, conflict-free b64 frag reads

typedef float v2f __attribute__((ext_vector_type(2)));
typedef float v8f __attribute__((ext_vector_type(8)));

__global__ void moe_init_ws(float* ws) {
    // zero probSum[64] + cntSum[64]
    if (threadIdx.x < 128) ws[threadIdx.x] = 0.0f;
}

// Async copy of 16B from global directly into LDS (ASYNCcnt-tracked, no VGPR staging).
__device__ __forceinline__ void async_copy_b128(void* lds_ptr, const float* gptr) {
    unsigned lds_addr = (unsigned)(uintptr_t)lds_ptr;  // low 32 bits of flat == LDS offset
    asm volatile("global_load_async_to_lds_b128 %0, %1, off"
                 :: "v"(lds_addr), "v"(gptr)
                 : "memory");
}

// Issue one chunk's A (128x64) + B (64x64) tiles: 12 async b128 ops per thread.
__device__ __forceinline__ void load_tiles_async(
    const float* __restrict__ hs, const float* __restrict__ gw,
    int tokenBase, int k0, float* aBuf, float* bBuf, int tid)
{
    #pragma unroll
    for (int i = 0; i < 8; ++i) {
        int idx = tid + i * 256;
        int row = idx >> 4;
        int c4  = (idx & 15) << 2;
        async_copy_b128(aBuf + row * PITCH + c4,
                        hs + (size_t)(tokenBase + row) * HIDDEN + k0 + c4);
    }
    #pragma unroll
    for (int i = 0; i < 4; ++i) {
        int idx = tid + i * 256;
        int row = idx >> 4;
        int c4  = (idx & 15) << 2;
        async_copy_b128(bBuf + row * PITCH + c4,
                        gw + (size_t)row * HIDDEN + k0 + c4);
    }
}

__global__ __launch_bounds__(256) void moe_router_kernel(
    const float* __restrict__ hs,      // [TOKENS][HIDDEN]
    const float* __restrict__ gw,      // [NEXP][HIDDEN]
    float* __restrict__ out_w,         // [TOKENS][2]
    float* __restrict__ out_idx,       // [TOKENS][2] (indices as float)
    float* __restrict__ probSum,       // [NEXP]
    float* __restrict__ cntSum)        // [NEXP]
{
    __shared__ float aT[2][MTILE * PITCH];  // 2 x 34816 B; aT[0] reused for logits
    __shared__ float bT[2][NEXP  * PITCH];  // 2 x 17408 B
    __shared__ float tMax[MTILE];
    __shared__ float tInv[MTILE];
    __shared__ float eAcc[NEXP];
    __shared__ float eCnt[NEXP];

    const int tid  = threadIdx.x;
    const int lane = tid & 31;
    const int wave = tid >> 5;
    const int l16  = lane & 15;
    const int half = lane >> 4;           // 0: K0/K1 slots, 1: K2/K3 slots
    const int wRow = wave * 16;           // this wave's token offset in tile
    const int tokenBase = blockIdx.x * MTILE;

    if (tid < NEXP) { eAcc[tid] = 0.0f; eCnt[tid] = 0.0f; }

    v8f c0 = {}, c1 = {}, c2 = {}, c3 = {};

    // --- prologue: start chunk 0 into buffer 0
    load_tiles_async(hs, gw, tokenBase, 0, aT[0], bT[0], tid);

    for (int c = 0; c < NCHUNK; ++c) {
        const int cur = c & 1;
        if (c + 1 < NCHUNK) {
            // overlap: start chunk c+1 into the other buffer (safe: last use of
            // that buffer was chunk c-1, fenced by the trailing barrier below)
            load_tiles_async(hs, gw, tokenBase, (c + 1) * KCHUNK,
                             aT[cur ^ 1], bT[cur ^ 1], tid);
            // 12 newer ops outstanding; chunk c's 12 ops retire in order
            asm volatile("s_wait_asynccnt 0xc" ::: "memory");
        } else {
            asm volatile("s_wait_asynccnt 0x0" ::: "memory");
        }
        __syncthreads();

        const float* aB = aT[cur];
        const float* bB = bT[cur];
        // --- WMMA f32 16x16x4 over the chunk: 16 k-steps x 4 N-tiles
        #pragma unroll
        for (int kk = 0; kk < KCHUNK; kk += 4) {
            const int ko = kk + 2 * half;
            v2f a  = *(const v2f*)(aB + (wRow + l16) * PITCH + ko);
            v2f b0 = *(const v2f*)(bB + ( 0 + l16) * PITCH + ko);
            v2f b1 = *(const v2f*)(bB + (16 + l16) * PITCH + ko);
            v2f b2 = *(const v2f*)(bB + (32 + l16) * PITCH + ko);
            v2f b3 = *(const v2f*)(bB + (48 + l16) * PITCH + ko);
            c0 = __builtin_amdgcn_wmma_f32_16x16x4_f32(false, a, false, b0, (short)0, c0, false, false);
            c1 = __builtin_amdgcn_wmma_f32_16x16x4_f32(false, a, false, b1, (short)0, c1, false, false);
            c2 = __builtin_amdgcn_wmma_f32_16x16x4_f32(false, a, false, b2, (short)0, c2, false, false);
            c3 = __builtin_amdgcn_wmma_f32_16x16x4_f32(false, a, false, b3, (short)0, c3, false, false);
        }
        __syncthreads();  // all waves done with buf[cur] before it is refilled
    }

    // --- spill logits to LDS (reuse aT[0]): row = local token, col = expert
    float* lgT = aT[0];
    #pragma unroll
    for (int v = 0; v < 8; ++v) {
        int row = wRow + half * 8 + v;
        lgT[row * PITCH +  0 + l16] = c0[v];
        lgT[row * PITCH + 16 + l16] = c1[v];
        lgT[row * PITCH + 32 + l16] = c2[v];
        lgT[row * PITCH + 48 + l16] = c3[v];
    }
    __syncthreads();

    // --- per-token softmax stats + top-2 (threads 0..127 each own a token)
    if (tid < MTILE) {
        const float* lg = lgT + tid * PITCH;
        float m = lg[0];
        #pragma unroll
        for (int e = 1; e < NEXP; ++e) m = fmaxf(m, lg[e]);
        float den = 0.0f;
        float p1 = -1.0f, p2 = -1.0f;
        int   i1 = 0,     i2 = 0;
        #pragma unroll
        for (int e = 0; e < NEXP; ++e) {
            float p = __expf(lg[e] - m);
            den += p;
            if (p > p1)      { p2 = p1; i2 = i1; p1 = p; i1 = e; }
            else if (p > p2) { p2 = p;  i2 = e; }
        }
        float inv = 1.0f / den;
        tMax[tid] = m;
        tInv[tid] = inv;
        float w1 = p1 * inv, w2 = p2 * inv;
        float s  = fmaxf(w1 + w2, 1e-8f);
        int g = tokenBase + tid;
        out_w[2 * g]     = w1 / s;
        out_w[2 * g + 1] = w2 / s;
        out_idx[2 * g]     = (float)i1;
        out_idx[2 * g + 1] = (float)i2;
        atomicAdd(&eCnt[i1], 1.0f);
    }
    __syncthreads();

    // --- per-expert prob-sum accumulation (expert-major to avoid atomic storms)
    {
        int e  = tid & 63;
        int t0 = (tid >> 6) * 32;
        float s = 0.0f;
        #pragma unroll
        for (int t = 0; t < 32; ++t) {
            int row = t0 + t;
            s += __expf(lgT[row * PITCH + e] - tMax[row]) * tInv[row];
        }
        atomicAdd(&eAcc[e], s);
    }
    __syncthreads();
    if (tid < NEXP) {
        atomicAdd(&probSum[tid], eAcc[tid]);
        atomicAdd(&cntSum[tid],  eCnt[tid]);
    }
}

__global__ void moe_finalize(const float* __restrict__ probSum,
                             const float* __restrict__ cntSum,
                             float* __restrict__ out_loss)
{
    __shared__ float red[NEXP];
    int e = threadIdx.x;
    float frac  = cntSum[e]  * (1.0f / (float)TOKENS);
    float meanp = probSum[e] * (1.0f / (float)TOKENS);
    red[e] = frac * meanp;
    __syncthreads();
    if (e == 0) {
        float s = 0.0f;
        #pragma unroll
        for (int i = 0; i < NEXP; ++i) s += red[i];
        *out_loss = (float)NEXP * s;
    }
}

extern "C" void kernel_launch(void* const* d_in, const int* in_sizes, int n_in,
                              void* d_out, int out_size, void* d_ws, size_t ws_size,
                              hipStream_t stream)
{
    const float* hs = (const float*)d_in[0];   // [32768][2048] f32
    const float* gw = (const float*)d_in[1];   // [64][2048] f32

    float* out      = (float*)d_out;
    float* out_w    = out;                      // 65536 floats
    float* out_idx  = out + (size_t)TOKENS * 2; // 65536 floats
    float* out_loss = out + (size_t)TOKENS * 4; // 1 float

    float* probSum = (float*)d_ws;              // 64 floats
    float* cntSum  = probSum + NEXP;            // 64 floats

    moe_init_ws<<<1, 128, 0, stream>>>(probSum);
    moe_router_kernel<<<TOKENS / MTILE, 256, 0, stream>>>(hs, gw, out_w, out_idx, probSum, cntSum);
    moe_finalize<<<1, NEXP, 0, stream>>>(probSum, cntSum, out_loss);
}